// Z1_Layer_start_43868795961568
// MI455X (gfx1250) — compile-verified
//
#include <hip/hip_runtime.h>
#include <hip/hip_bf16.h>

typedef __attribute__((ext_vector_type(2))) float v2f;
typedef __attribute__((ext_vector_type(8))) float v8f;

#define WMMA_BLOCKS 225                       // 72 slices * 25 tiles = 1800 waves / 8 waves-per-block
#define A2_ELEMS    (13 * 4900)               // 63700
#define FILL_ELEMS  (A2_ELEMS + 85 + 4900)    // 68685
#define FILL_BLOCKS ((FILL_ELEMS + 255) / 256) // 269
#define AA_ELEMS    (85 * 4900)               // 416500
#define B_OFF       AA_ELEMS                  // 416500
#define A0_OFF      (AA_ELEMS + 85)           // 416585

// U[e, m] entry of uv(off, corner, h): (m == off+e) + (m == corner)*h[e]
__device__ __forceinline__ float uelem(int m, int off_e, int cor, float h) {
    float v = (m == off_e) ? 1.0f : 0.0f;
    if (m == cor) v += h;
    return v;
}

__global__ void __launch_bounds__(256)
build_all(const float* __restrict__ H1, const float* __restrict__ v1,
          const float* __restrict__ t,  const float* __restrict__ c,
          float* __restrict__ out)
{
    if (blockIdx.x < WMMA_BLOCKS) {
        // ================= Asvec / Bsvec via V_WMMA_F32_16X16X4_F32 =================
        const int wid   = blockIdx.x * 8 + (threadIdx.x >> 5); // wave id, 0..1799
        const int lane  = threadIdx.x & 31;
        const int slice = wid / 25;            // 0..71  (0..35 -> A, 36..71 -> B)
        const int tile  = wid % 25;
        const int tm = tile / 5, tn = tile % 5;
        const bool isB  = slice >= 36;
        // triu_indices(8) pair for this slice (wave-uniform)
        int q = isB ? slice - 36 : slice;
        int i = 0, rowlen = 8;
        while (q >= rowlen) { q -= rowlen; --rowlen; ++i; }
        const int j = i + q;

        // h columns: h111=0.2*H1[:,0], h112=0.2*H1[:,2], h121=0.2*H1[:,5], h122=0.2*H1[:,7]
        const float hi0 = 0.2f * H1[i*8 + 0], hi1 = 0.2f * H1[i*8 + 2];
        const float hi2 = 0.2f * H1[i*8 + 5], hi3 = 0.2f * H1[i*8 + 7];
        const float hj0 = 0.2f * H1[j*8 + 0], hj1 = 0.2f * H1[j*8 + 2];
        const float hj2 = 0.2f * H1[j*8 + 5], hj3 = 0.2f * H1[j*8 + 7];
        // coefficient vector (b = 0..3); b=4 handled as delta-fixup below
        const float c0 = isB ? -1.0f : 0.1f;
        const float c1 = isB ?  0.1f : -1.0f;
        const float c2 = -1.0f, c3 = -1.0f;

        // lane-half K split: lanes 0-15 hold K={0,1}, lanes 16-31 hold K={2,3}
        const bool hh = (lane >= 16);
        const int  mA = tm * 16 + (lane & 15);   // A-fragment row (M)
        const int  nB = tn * 16 + (lane & 15);   // B-fragment col (N)
        const int offx = hh ? 18 : 0,  offy = hh ? 27 : 9;
        const int corx = hh ? 26 : 8,  cory = hh ? 35 : 17;
        const float cx  = hh ? c2  : c0,  cy  = hh ? c3  : c1;
        const float hix = hh ? hi2 : hi0, hiy = hh ? hi3 : hi1;
        const float hjx = hh ? hj2 : hj0, hjy = hh ? hj3 : hj1;

        // WMMA #1: D += (cA .* UA[:,i,m])^T * UA[:,j,n]
        v2f a1, b1, a2, b2;
        a1.x = cx * uelem(mA, offx + i, corx, hix);
        a1.y = cy * uelem(mA, offy + i, cory, hiy);
        b1.x =      uelem(nB, offx + j, corx, hjx);
        b1.y =      uelem(nB, offy + j, cory, hjy);
        // WMMA #2 (transpose term, i<->j): D += (cA .* UA[:,j,m])^T * UA[:,i,n]
        a2.x = cx * uelem(mA, offx + j, corx, hjx);
        a2.y = cy * uelem(mA, offy + j, cory, hjy);
        b2.x =      uelem(nB, offx + i, corx, hix);
        b2.y =      uelem(nB, offy + i, cory, hiy);

        v8f acc = {};
        acc = __builtin_amdgcn_wmma_f32_16x16x4_f32(false, a1, false, b1,
                                                    (short)0, acc, false, false);
        acc = __builtin_amdgcn_wmma_f32_16x16x4_f32(false, a2, false, b2,
                                                    (short)0, acc, false, false);

        // fixups: b=4 rank-1 delta term, doubled I8(x)D6869 diag term, triu scale, negate
        const int   c4    = isB ? 44 : 36;        // off of uv for b=4 (corner has h=0)
        const float scale = (i == j) ? 0.5f : 1.0f;
        const int   half8 = hh ? 8 : 0;
        const long  slice_base = (long)slice * 4900;
        #pragma unroll
        for (int r = 0; r < 8; ++r) {
            const int m = tm * 16 + r + half8;    // C/D layout: VGPR r -> M = r (+8 hi half)
            const int n = nB;
            float v = acc[r];
            if (m == c4 + i && n == c4 + j) v += 1.0f;
            if (m == c4 + j && n == c4 + i) v += 1.0f;
            if (i == j && ((m == 68 && n == 69) || (m == 69 && n == 68))) v += 2.0f;
            v = -(v * scale);
            if (m < 70 && n < 70) out[slice_base + m * 70 + n] = v;
        }
    } else {
        // ================= A2 slices, b vector, A0 (structured fill) =================
        const int idx = (blockIdx.x - WMMA_BLOCKS) * 256 + threadIdx.x;
        if (idx >= FILL_ELEMS) return;
        if (idx < A2_ELEMS) {
            const int k2  = idx / 4900;
            const int rem = idx % 4900;
            const int m = rem / 70, n = rem % 70;
            float v = 0.0f;
            if (k2 < 4) {
                const int s = 9 * k2;
                if (m == n) {
                    if (m >= s && m < s + 8)   v = 400.0f;   // Q * I8 block
                    else if (m == s + 8)       v = -1.0f;
                    else if (m == 52 + k2)     v = 1.0f;
                }
            } else if (k2 < 8) {
                const int t4  = k2 - 4;
                const int cnr = 8 + 9 * t4;                  // 8,17,26,35
                if (m == n) {
                    if (m == cnr)              v = (t4 < 2) ? -1.0f : 1.0f;
                    else if (m == 56 + t4)     v = 1.0f;
                }
                const int r0 = 60 + 2 * t4;
                if ((m == r0 && n == r0 + 1) || (m == r0 + 1 && n == r0)) v = 1.0f;
            } else {
                const int d = 60 + 2 * (k2 - 8);             // 60,62,64,66,68
                if (m == n && m == d) v = 1.0f;
            }
            out[72 * 4900 + idx] = -v;                       // AA = -concat(...)
        } else if (idx < A2_ELEMS + 85) {
            const int r = idx - A2_ELEMS;
            float bv = 0.0f;
            if (r== 0||r== 8||r==15||r==21||r==26||r==30||r==33||r==35||
                r==36||r==44||r==51||r==57||r==62||r==66||r==69||r==71)
                bv = -1.0f;
            else if (r >= 76 && r < 80) bv = -v1[r - 76];
            else if (r >= 80)           bv = -0.5f * c[0];
            out[B_OFF + r] = bv;                             // second output is -b
        } else {
            const int e = idx - (A2_ELEMS + 85);
            const int m = e / 70, n = e % 70;
            float v = 0.0f;
            if (m == n) {
                if (m == 8 || m == 17)                              v = -0.0004f; // -sigma2
                else if (m==61 || m==63 || m==65 || m==67 || m==69) v = 1.0f;
            } else if (m >= 60 && m <= 67 && (m >> 1) == (n >> 1)) {
                const int pidx = (m >> 1) - 30;              // 0..3 for pairs (60,61)..(66,67)
                const int tidx = (pidx < 2) ? pidx + 2 : pidx - 2; // tv = t[[2,3,0,1]]
                v = t[tidx];
            }
            out[A0_OFF + e] = v;                             // A0 is NOT negated
        }
    }
}

extern "C" void kernel_launch(void* const* d_in, const int* in_sizes, int n_in,
                              void* d_out, int out_size, void* d_ws, size_t ws_size,
                              hipStream_t stream) {
    // setup_inputs order: rho1(0), mu1(1), v1(2), t(3), H1(4), c(5); rho1/mu1 unused.
    const float* v1 = (const float*)d_in[2];
    const float* t  = (const float*)d_in[3];
    const float* H1 = (const float*)d_in[4];
    const float* c  = (const float*)d_in[5];
    build_all<<<WMMA_BLOCKS + FILL_BLOCKS, 256, 0, stream>>>(H1, v1, t, c, (float*)d_out);
}